// GCN_6_3_leaky_relu_8332236554735
// MI455X (gfx1250) — compile-verified
//
#include <hip/hip_runtime.h>
#include <math.h>

// ---------------------------------------------------------------------------
// GCN 6-layer pipeline for gfx1250 (MI455X), wave32.
//   per layer: zero agg -> edge scatter-add (f32 atomics, L2-resident)
//              -> WMMA f32 16x16x4 GEMM (+bias), branch-free inner loop
//              -> activation (leaky-relu / softmax / log-softmax)
// ---------------------------------------------------------------------------

typedef __attribute__((ext_vector_type(2))) float v2f;
typedef __attribute__((ext_vector_type(8))) float v8f;

__global__ void fill_zero_kernel(float* __restrict__ p, long long n) {
    long long i = (long long)blockIdx.x * blockDim.x + threadIdx.x;
    long long stride = (long long)gridDim.x * blockDim.x;
    for (; i < n; i += stride) p[i] = 0.0f;
}

// One wave per edge; lanes stride the feature dimension -> coalesced gather
// from x[src[e],:] and coalesced f32 atomic-add stream into agg[dst[e],:].
// Feature matrix fits in the 192MB L2, so both sides stay on-chip.
__global__ __launch_bounds__(256)
void scatter_add_kernel(const float* __restrict__ x,
                        const int* __restrict__ src,
                        const int* __restrict__ dst,
                        float* __restrict__ agg,
                        int E, int F) {
    int wave = (int)((blockIdx.x * blockDim.x + threadIdx.x) >> 5);
    int lane = threadIdx.x & 31;
    if (wave >= E) return;                       // whole-wave uniform exit
    const float* xr = x + (long long)src[wave] * F;
    float* ar = agg + (long long)dst[wave] * F;
    for (int f = lane; f < F; f += 32)
        atomicAdd(&ar[f], xr[f]);
}

// Y[N,Fout] = A[N,K] @ W[K,Fout] + bias, via V_WMMA_F32_16X16X4_F32.
// One wave computes one 16x16 output tile, stepping K by 4.
// A layout (16x4 f32): lane -> M = lane%16; half-wave (lane>=16) -> K base +2.
// B layout (4x16 f32): lane -> Ncol = lane%16; same K split.
// C/D layout: lane%16 = column, VGPR v -> row v + 8*(lane/16).
//
// Branch-free body: out-of-range rows/columns load from CLAMPED indices
// (garbage feeds only D elements that are never stored), so EXEC stays
// all-1s with no saveexec branching in the K loop. The K%4 tail uses
// clamped addresses + value selects (v_cndmask), still branch-free.
__global__ __launch_bounds__(256)
void gemm_bias_wmma_kernel(const float* __restrict__ A,
                           const float* __restrict__ W,
                           const float* __restrict__ bias,
                           float* __restrict__ Y,
                           int Nrows, int K, int Fout,
                           int tilesM, int tilesN) {
    int wave = (int)(blockIdx.x * (blockDim.x >> 5) + (threadIdx.x >> 5));
    if (wave >= tilesM * tilesN) return;         // whole-wave uniform exit
    int lane = threadIdx.x & 31;
    int mt = wave / tilesN;
    int nt = wave - mt * tilesN;
    int m0 = mt << 4, n0 = nt << 4;
    int half = lane >> 4;                         // 0: K=0,1  1: K=2,3
    int l16  = lane & 15;
    int m = m0 + l16;                             // A row for this lane
    int n = n0 + l16;                             // B/D column for this lane
    bool nvalid = n < Fout;
    int mc = (m < Nrows) ? m : (Nrows - 1);       // clamped load indices
    int nc = nvalid ? n : (Fout - 1);
    const float* __restrict__ Arow = A + (long long)mc * K;
    const float* __restrict__ Wcol = W + nc;

    v8f acc = {};
    const int K4 = K & ~3;
    #pragma unroll 4
    for (int k0 = 0; k0 < K4; k0 += 4) {
        int ka = k0 + half * 2;
        v2f a, b;
        a.x = Arow[ka];
        a.y = Arow[ka + 1];
        b.x = Wcol[(long long)ka * Fout];
        b.y = Wcol[(long long)(ka + 1) * Fout];
        acc = __builtin_amdgcn_wmma_f32_16x16x4_f32(
            false, a, false, b, (short)0, acc, false, false);
    }
    if (K4 < K) {                                 // tail: K%4 in {1,2,3}
        int ka  = K4 + half * 2;
        int k0c = (ka     < K) ? ka     : (K - 1);
        int k1c = (ka + 1 < K) ? ka + 1 : (K - 1);
        float a0 = Arow[k0c], a1 = Arow[k1c];
        float b0 = Wcol[(long long)k0c * Fout];
        float b1 = Wcol[(long long)k1c * Fout];
        v2f a, b;
        a.x = (ka     < K) ? a0 : 0.0f;
        a.y = (ka + 1 < K) ? a1 : 0.0f;
        b.x = (ka     < K) ? b0 : 0.0f;
        b.y = (ka + 1 < K) ? b1 : 0.0f;
        acc = __builtin_amdgcn_wmma_f32_16x16x4_f32(
            false, a, false, b, (short)0, acc, false, false);
    }

    float bv = bias[nc];
    #pragma unroll
    for (int v = 0; v < 8; ++v) {
        int mm = m0 + v + 8 * half;
        if (mm < Nrows && nvalid)
            Y[(long long)mm * Fout + n] = acc[v] + bv;
    }
}

__global__ void leaky_relu_kernel(float* __restrict__ y, long long n) {
    long long i = (long long)blockIdx.x * blockDim.x + threadIdx.x;
    long long stride = (long long)gridDim.x * blockDim.x;
    for (; i < n; i += stride) {
        float v = y[i];
        y[i] = v > 0.0f ? v : 0.01f * v;
    }
}

// One wave per row; shuffle reductions over 32 lanes (wave32).
// logmode=0: softmax; logmode=1: log_softmax. In-place.
__global__ __launch_bounds__(256)
void softmax_kernel(float* __restrict__ y, int N, int F, int logmode) {
    int row = (int)((blockIdx.x * blockDim.x + threadIdx.x) >> 5);
    int lane = threadIdx.x & 31;
    if (row >= N) return;                        // whole-wave uniform exit
    float* r = y + (long long)row * F;

    float m = -INFINITY;
    for (int f = lane; f < F; f += 32) m = fmaxf(m, r[f]);
    #pragma unroll
    for (int o = 16; o > 0; o >>= 1) m = fmaxf(m, __shfl_xor(m, o, 32));

    float s = 0.0f;
    for (int f = lane; f < F; f += 32) s += __expf(r[f] - m);
    #pragma unroll
    for (int o = 16; o > 0; o >>= 1) s += __shfl_xor(s, o, 32);

    if (logmode) {
        float ls = __logf(s);
        for (int f = lane; f < F; f += 32) r[f] = r[f] - m - ls;
    } else {
        float inv = 1.0f / s;
        for (int f = lane; f < F; f += 32) r[f] = __expf(r[f] - m) * inv;
    }
}

extern "C" void kernel_launch(void* const* d_in, const int* in_sizes, int n_in,
                              void* d_out, int out_size, void* d_ws, size_t ws_size,
                              hipStream_t stream) {
    const int dims[7] = {300, 200, 150, 100, 50, 25, 3};
    const float* features = (const float*)d_in[0];
    const int*   src      = (const int*)d_in[1];
    const int*   dst      = (const int*)d_in[2];
    const float* Wt[6];
    const float* bs[6];
    for (int l = 0; l < 6; ++l) {
        Wt[l] = (const float*)d_in[3 + 2 * l];
        bs[l] = (const float*)d_in[4 + 2 * l];
    }
    const int N = in_sizes[0] / dims[0];   // 50000
    const int E = in_sizes[1];             // 262144

    // Workspace: agg (N*300) | ping (N*200) | pong (N*200)  ~= 140 MB
    float* agg  = (float*)d_ws;
    float* bufB = agg  + (long long)N * 300;
    float* bufC = bufB + (long long)N * 200;

    const float* x = features;
    for (int l = 0; l < 6; ++l) {
        const int Fin  = dims[l];
        const int Fout = dims[l + 1];

        // 1) agg = 0
        long long nAgg = (long long)N * Fin;
        fill_zero_kernel<<<4096, 256, 0, stream>>>(agg, nAgg);

        // 2) agg[dst[e],:] += x[src[e],:]  (wave per edge)
        int scatterBlocks = (int)(((long long)E * 32 + 255) / 256);
        scatter_add_kernel<<<scatterBlocks, 256, 0, stream>>>(x, src, dst, agg, E, Fin);

        // 3) y = agg @ W + b  (WMMA f32 16x16x4)
        float* y = (l == 5) ? (float*)d_out : ((l & 1) ? bufC : bufB);
        int tilesM = (N + 15) / 16;
        int tilesN = (Fout + 15) / 16;
        int waves  = tilesM * tilesN;
        int gemmBlocks = (waves + 7) / 8;   // 8 waves / 256-thread block
        gemm_bias_wmma_kernel<<<gemmBlocks, 256, 0, stream>>>(
            agg, Wt[l], bs[l], y, N, Fin, Fout, tilesM, tilesN);

        // 4) activation
        if (l == 0 || l == 2 || l == 4) {
            leaky_relu_kernel<<<4096, 256, 0, stream>>>(y, (long long)N * Fout);
        } else {
            int smBlocks = (int)(((long long)N * 32 + 255) / 256);
            softmax_kernel<<<smBlocks, 256, 0, stream>>>(y, N, Fout, (l == 5) ? 1 : 0);
        }
        x = y;
    }
}